// GumbelSinkhorn_25443386261686
// MI455X (gfx1250) — compile-verified
//
#include <hip/hip_runtime.h>

#define DIM 4096
#define N_ITERS 20
#define TILE_ROWS 128
#define ROW_TILES (DIM / TILE_ROWS)   // 32

typedef __attribute__((ext_vector_type(2))) float v2f;
typedef __attribute__((ext_vector_type(8))) float v8f;

__device__ __forceinline__ float gumbel_from_uniform(float u) {
  const float EPS = 1e-10f;
  return -__logf(-__logf(u + EPS) + EPS);
}

// One block (256 threads) per row. Each thread holds 16 elements in registers.
// FIRST: src0 = logits, src1 = noise  -> la = rownorm(logits + gumbel(noise))
// else : src0 = la,     src1 = colLSE -> la = rownorm(la - colLSE[col])
template <bool FIRST>
__global__ __launch_bounds__(256)
void row_norm_kernel(const float* __restrict__ src0,
                     const float* __restrict__ src1,
                     float* __restrict__ la)
{
  const int row  = blockIdx.x;
  const int tid  = threadIdx.x;
  const int lane = tid & 31;
  const int wave = tid >> 5;

  const float4* rowp = (const float4*)(src0 + (size_t)row * DIM);
  float4 x[4];
#pragma unroll
  for (int i = 0; i < 4; ++i) {
    const int c4 = tid + 256 * i;            // float4 index in row (coalesced)
    float4 v = rowp[c4];
    if (FIRST) {
      const float4* np = (const float4*)(src1 + (size_t)row * DIM);
      float4 u = np[c4];
      v.x += gumbel_from_uniform(u.x);
      v.y += gumbel_from_uniform(u.y);
      v.z += gumbel_from_uniform(u.z);
      v.w += gumbel_from_uniform(u.w);
    } else {
      float4 cl = ((const float4*)src1)[c4];
      v.x -= cl.x; v.y -= cl.y; v.z -= cl.z; v.w -= cl.w;
    }
    x[i] = v;
  }

  // ---- block max (wave32 shuffle tree + LDS combine) ----
  float m = -INFINITY;
#pragma unroll
  for (int i = 0; i < 4; ++i)
    m = fmaxf(m, fmaxf(fmaxf(x[i].x, x[i].y), fmaxf(x[i].z, x[i].w)));
#pragma unroll
  for (int off = 16; off > 0; off >>= 1)
    m = fmaxf(m, __shfl_xor(m, off, 32));

  __shared__ float wred[8];
  __shared__ float parts[256];
  __shared__ float sS;
  if (lane == 0) wred[wave] = m;
  __syncthreads();
  float M = wred[0];
#pragma unroll
  for (int w = 1; w < 8; ++w) M = fmaxf(M, wred[w]);

  // ---- per-thread sum of exp (registers only) ----
  float p = 0.f;
#pragma unroll
  for (int i = 0; i < 4; ++i) {
    p += __expf(x[i].x - M);
    p += __expf(x[i].y - M);
    p += __expf(x[i].z - M);
    p += __expf(x[i].w - M);
  }
  parts[tid] = p;
  __syncthreads();

  // ---- block sum of 256 partials via V_WMMA_F32_16X16X4_F32 ----
  // parts[] row-major as a 16x16 tile P. f32 B(4x16) layout: VGPR0 lanes0-15 = K=0,
  // lanes16-31 = K=1; VGPR1 = K=2,3  =>  b.x = parts[64r + lane], b.y = parts[64r + 32 + lane].
  // A = ones(16x4) => D[m][n] = colsum(P)[n] for all m; accumulate over the 4 K-chunks.
  if (tid < 32) {                    // full wave 0, EXEC all ones
    v8f c = {};
    const v2f a = {1.0f, 1.0f};
#pragma unroll
    for (int r = 0; r < 4; ++r) {
      v2f b;
      b.x = parts[64 * r + tid];
      b.y = parts[64 * r + 32 + tid];
      c = __builtin_amdgcn_wmma_f32_16x16x4_f32(false, a, false, b,
                                                (short)0, c, false, false);
    }
    float s = c[0];                  // lane n(<16): colsum[n]; lanes 16-31 replicate
    s += __shfl_xor(s, 1, 32);
    s += __shfl_xor(s, 2, 32);
    s += __shfl_xor(s, 4, 32);
    s += __shfl_xor(s, 8, 32);       // lanes 0-15 group now holds total sum
    if (tid == 0) sS = s;
  }
  __syncthreads();

  const float lse = M + __logf(sS);
  float4* outp = (float4*)(la + (size_t)row * DIM);
#pragma unroll
  for (int i = 0; i < 4; ++i) {
    float4 v = x[i];
    v.x -= lse; v.y -= lse; v.z -= lse; v.w -= lse;
    outp[tid + 256 * i] = v;
  }
}

// Per-column partial (max, sum of exp) over a 128-row tile. Data is L2-resident,
// so the second pass re-reads cheaply; one exp per element.
__global__ __launch_bounds__(256)
void col_partial_kernel(const float* __restrict__ la,
                        float* __restrict__ pmax,
                        float* __restrict__ psum)
{
  const int col = blockIdx.x * 256 + threadIdx.x;
  const int r0  = blockIdx.y * TILE_ROWS;
  const float* base = la + (size_t)r0 * DIM + col;

  float m = -INFINITY;
#pragma unroll 4
  for (int r = 0; r < TILE_ROWS; ++r) {
    if (r + 16 < TILE_ROWS)
      __builtin_prefetch(base + (size_t)(r + 16) * DIM, 0, 0);  // global_prefetch_b8
    m = fmaxf(m, base[(size_t)r * DIM]);
  }
  float s = 0.f;
#pragma unroll 4
  for (int r = 0; r < TILE_ROWS; ++r)
    s += __expf(base[(size_t)r * DIM] - m);

  pmax[blockIdx.y * DIM + col] = m;
  psum[blockIdx.y * DIM + col] = s;
}

// Merge the 32 per-tile partials into colLSE[col] = M + log(sum_i s_i * exp(m_i - M)).
__global__ __launch_bounds__(256)
void col_finalize_kernel(const float* __restrict__ pmax,
                         const float* __restrict__ psum,
                         float* __restrict__ colLSE)
{
  const int col = blockIdx.x * 256 + threadIdx.x;
  float M = -INFINITY;
#pragma unroll
  for (int t = 0; t < ROW_TILES; ++t)
    M = fmaxf(M, pmax[t * DIM + col]);
  float S = 0.f;
#pragma unroll
  for (int t = 0; t < ROW_TILES; ++t)
    S += psum[t * DIM + col] * __expf(pmax[t * DIM + col] - M);
  colLSE[col] = M + __logf(S);
}

// out = exp(la - colLSE[col]), in place over d_out.
__global__ __launch_bounds__(256)
void final_exp_kernel(float* __restrict__ la, const float* __restrict__ colLSE)
{
  const size_t i4  = (size_t)blockIdx.x * 256 + threadIdx.x;  // float4 index
  const int    c4  = (int)(i4 & (DIM / 4 - 1));
  float4 v  = ((float4*)la)[i4];
  float4 cl = ((const float4*)colLSE)[c4];
  v.x = __expf(v.x - cl.x);
  v.y = __expf(v.y - cl.y);
  v.z = __expf(v.z - cl.z);
  v.w = __expf(v.w - cl.w);
  ((float4*)la)[i4] = v;
}

extern "C" void kernel_launch(void* const* d_in, const int* in_sizes, int n_in,
                              void* d_out, int out_size, void* d_ws, size_t ws_size,
                              hipStream_t stream) {
  const float* logits = (const float*)d_in[0];
  const float* noise  = (const float*)d_in[1];
  float* la = (float*)d_out;                 // 64 MB working buffer (L2-resident)

  float* colLSE = (float*)d_ws;              // DIM floats
  float* pmax   = colLSE + DIM;              // ROW_TILES*DIM floats
  float* psum   = pmax + ROW_TILES * DIM;    // ROW_TILES*DIM floats

  // Iteration 1 row-normalize, fused with Gumbel init.
  row_norm_kernel<true><<<DIM, 256, 0, stream>>>(logits, noise, la);

  for (int t = 0; t < N_ITERS; ++t) {
    col_partial_kernel<<<dim3(DIM / 256, ROW_TILES), 256, 0, stream>>>(la, pmax, psum);
    col_finalize_kernel<<<DIM / 256, 256, 0, stream>>>(pmax, psum, colLSE);
    if (t < N_ITERS - 1) {
      // Next iteration's row-normalize, fused with this iteration's col subtract.
      row_norm_kernel<false><<<DIM, 256, 0, stream>>>(la, colLSE, la);
    }
  }
  // Last col subtract fused with the final exp.
  final_exp_kernel<<<(DIM * DIM / 4) / 256, 256, 0, stream>>>(la, colLSE);
}